// BNN_KDE_52510270161354
// MI455X (gfx1250) — compile-verified
//
#include <hip/hip_runtime.h>
#include <math.h>

#define K_COMP 8192
#define N_SAMP 8192
#define B_X    2048
#define DW     13
#define LOG_2PI 1.8378770664093453f
#define BETA    5.0f
#define LOG_K   9.010913347279288f   // log(8192)

typedef __attribute__((ext_vector_type(2))) float v2f;
typedef __attribute__((ext_vector_type(8))) float v8f;

// workspace layout (float offsets)
#define WS_EMP16   0                  // 8192*16
#define WS_W16     131072             // 8192*16
#define WS_CPACK   262144             // 8192*2 (float2: {1/var, C_j})
#define WS_SSW     278528             // 8192
#define WS_PRIOR   286720             // 8192
#define WS_DATAP   294912             // 8192  per-sample data partials
#define WS_KLP     303104             // 512   per-wave KL partials
// total 303616 floats = ~1.19 MB

__global__ __launch_bounds__(256)
void k_prep(const float* __restrict__ emp, const float* __restrict__ rhos,
            const float* __restrict__ eps, const int* __restrict__ idxs,
            float* __restrict__ ws)
{
    int t = blockIdx.x * 256 + threadIdx.x;
    if (t >= K_COMP) return;

    // component role: pad emp row, pack {1/var, C_j}
    float rho  = rhos[t];
    float stdc = log1pf(__expf(rho));          // softplus
    float var  = stdc * stdc;
    float rinv = 1.0f / var;
    float logv = __logf(var);
    float sse  = 0.0f;
    float* e16 = ws + WS_EMP16 + t * 16;
    #pragma unroll
    for (int d = 0; d < DW; ++d) { float v = emp[t * DW + d]; e16[d] = v; sse = fmaf(v, v, sse); }
    e16[13] = 0.0f; e16[14] = 0.0f; e16[15] = 0.0f;
    float2* cpk = (float2*)(ws + WS_CPACK);
    cpk[t] = make_float2(rinv, fmaf(sse, rinv, (float)DW * (LOG_2PI + logv)));

    // sample role: build padded w row, |w|^2, prior_lp
    int   j    = idxs[t];
    float stdj = log1pf(__expf(rhos[j]));
    float ssw  = 0.0f;
    float* w16 = ws + WS_W16 + t * 16;
    #pragma unroll
    for (int d = 0; d < DW; ++d) {
        float wv = fmaf(eps[t * DW + d], stdj, emp[j * DW + d]);
        w16[d] = wv; ssw = fmaf(wv, wv, ssw);
    }
    w16[13] = 0.0f; w16[14] = 0.0f; w16[15] = 0.0f;
    ws[WS_SSW   + t] = ssw;
    ws[WS_PRIOR + t] = -0.5f * ssw - 0.5f * (float)DW * LOG_2PI;  // ALPHA=1
}

__global__ __launch_bounds__(256)
void k_reg(const float* __restrict__ x, const float* __restrict__ y,
           float* __restrict__ ws)
{
    int wave = (blockIdx.x * 256 + threadIdx.x) >> 5;   // one wave per sample
    int lane = threadIdx.x & 31;
    const float* w = ws + WS_W16 + wave * 16;
    float w10 = w[0],  w11 = w[1],  b10 = w[2], b11 = w[3];
    float w200 = w[4], w201 = w[5], w210 = w[6], w211 = w[7];
    float b20 = w[8],  b21 = w[9];
    float w30 = w[10], w31 = w[11], b3 = w[12];

    float acc = 0.0f;
    for (int k = lane; k < B_X; k += 32) {
        float xv = x[k];
        float h0 = tanhf(fmaf(w10, xv, b10));
        float h1 = tanhf(fmaf(w11, xv, b11));
        float g0 = tanhf(fmaf(w200, h0, fmaf(w201, h1, b20)));
        float g1 = tanhf(fmaf(w210, h0, fmaf(w211, h1, b21)));
        float yp = fmaf(w30, g0, fmaf(w31, g1, b3));
        float e  = yp - y[k];
        acc = fmaf(e, e, acc);
    }
    #pragma unroll
    for (int off = 16; off >= 1; off >>= 1) acc += __shfl_xor(acc, off, 32);
    if (lane == 0) ws[WS_DATAP + wave] = -0.5f * BETA * acc;
}

__global__ __launch_bounds__(128)
void k_kde(float* __restrict__ ws)
{
    const float*  e16 = ws + WS_EMP16;
    const float*  w16 = ws + WS_W16;
    const float2* cpk = (const float2*)(ws + WS_CPACK);

    int wave  = (blockIdx.x * 128 + threadIdx.x) >> 5;   // 512 waves, one per 16-row tile
    int lane  = threadIdx.x & 31;
    int rbase = wave * 16;
    int m     = lane & 15;           // A: row, B: col, C/D: col
    int kb    = (lane >> 4) * 2;     // K base within 4-chunk for this lane-half
    int h8    = (lane >> 4) * 8;     // C/D row offset for this lane-half

    // A tile (rows), pre-scaled by -2 so WMMA accumulates -2*dot
    v2f a[4];
    #pragma unroll
    for (int c = 0; c < 4; ++c) {
        const v2f t = *(const v2f*)(w16 + (rbase + m) * 16 + 4 * c + kb);
        a[c] = t * -2.0f;
    }
    float ssw[8];
    #pragma unroll
    for (int v = 0; v < 8; ++v) ssw[v] = ws[WS_SSW + rbase + h8 + v];

    float mx[8], sm[8];
    #pragma unroll
    for (int v = 0; v < 8; ++v) { mx[v] = -3.0e38f; sm[v] = 0.0f; }

    for (int jt = 0; jt < K_COMP / 16; ++jt) {
        int col = jt * 16 + m;
        v2f b[4];
        #pragma unroll
        for (int c = 0; c < 4; ++c)
            b[c] = *(const v2f*)(e16 + col * 16 + 4 * c + kb);
        float2 cp = cpk[col];

        v8f acc = {};
        #pragma unroll
        for (int c = 0; c < 4; ++c)
            acc = __builtin_amdgcn_wmma_f32_16x16x4_f32(
                false, a[c], false, b[c], (short)0, acc, false, false);

        // comp_lp = -0.5*(C_j + (|w_i|^2 - 2*dot) / var_j); streaming logsumexp
        #pragma unroll
        for (int v = 0; v < 8; ++v) {
            float val = -0.5f * fmaf(ssw[v] + acc[v], cp.x, cp.y);
            float mn  = fmaxf(mx[v], val);
            sm[v] = fmaf(sm[v], __expf(mx[v] - mn), __expf(val - mn));
            mx[v] = mn;
        }
    }

    // combine the 16 column-slots within each lane-half (butterfly, width 16)
    #pragma unroll
    for (int off = 8; off >= 1; off >>= 1) {
        #pragma unroll
        for (int v = 0; v < 8; ++v) {
            float mo = __shfl_xor(mx[v], off, 16);
            float so = __shfl_xor(sm[v], off, 16);
            float mn = fmaxf(mx[v], mo);
            sm[v] = fmaf(sm[v], __expf(mx[v] - mn), so * __expf(mo - mn));
            mx[v] = mn;
        }
    }

    // every lane in a half now holds the full (m,s); compute KL partial
    float kl = 0.0f;
    #pragma unroll
    for (int v = 0; v < 8; ++v) {
        int r = rbase + h8 + v;
        float q = mx[v] + __logf(sm[v]) - LOG_K;
        kl += q - ws[WS_PRIOR + r];
    }
    kl += __shfl_xor(kl, 16, 32);          // add the two lane-halves (rows 0-7 + 8-15)
    kl *= (1.0f / 16.0f);                  // each half's value replicated across 16 lanes
    if (lane == 0) ws[WS_KLP + wave] = kl;
}

__global__ __launch_bounds__(256)
void k_fin(const float* __restrict__ ws, float* __restrict__ out)
{
    __shared__ float sd[256], sk[256];
    int t = threadIdx.x;
    float ad = 0.0f, ak = 0.0f;
    for (int i = t; i < N_SAMP; i += 256) ad += ws[WS_DATAP + i];
    for (int i = t; i < 512;    i += 256) ak += ws[WS_KLP + i];
    sd[t] = ad; sk[t] = ak; __syncthreads();
    for (int o = 128; o > 0; o >>= 1) {
        if (t < o) { sd[t] += sd[t + o]; sk[t] += sk[t + o]; }
        __syncthreads();
    }
    if (t == 0) {
        float data_lp = sd[0] / (float)N_SAMP
                      + (float)B_X * 0.5f * (__logf(BETA) - LOG_2PI);
        float kl_term = sk[0] / (float)N_SAMP;
        out[0] = data_lp - kl_term;    // KL_BETA = 1
    }
}

extern "C" void kernel_launch(void* const* d_in, const int* in_sizes, int n_in,
                              void* d_out, int out_size, void* d_ws, size_t ws_size,
                              hipStream_t stream)
{
    const float* emp  = (const float*)d_in[0];   // (8192,13)
    const float* rhos = (const float*)d_in[1];   // (8192,)
    const float* x    = (const float*)d_in[2];   // (2048,1)
    const float* y    = (const float*)d_in[3];   // (2048,)
    const float* eps  = (const float*)d_in[4];   // (8192,13)
    const int*   idxs = (const int*)d_in[5];     // (8192,)
    float* ws  = (float*)d_ws;
    float* out = (float*)d_out;

    k_prep<<<K_COMP / 256, 256, 0, stream>>>(emp, rhos, eps, idxs, ws);
    k_reg <<<N_SAMP / 8,  256, 0, stream>>>(x, y, ws);          // 8 waves/block, 1 wave/sample
    k_kde <<<(N_SAMP / 16) / 4, 128, 0, stream>>>(ws);          // 4 waves/block, 1 wave/16 rows
    k_fin <<<1, 256, 0, stream>>>(ws, out);
}